// MGATLayer_17025250362059
// MI455X (gfx1250) — compile-verified
//
#include <hip/hip_runtime.h>
#include <hip/hip_bf16.h>

#define IN_CH 64
#define HEADS 4
#define OUT_CH 16
#define NEG_SLOPE 0.2f

typedef __attribute__((ext_vector_type(2))) float v2f;
typedef __attribute__((ext_vector_type(8))) float v8f;

__device__ __forceinline__ unsigned enc_f(float f) {
    unsigned u = __float_as_uint(f);
    return (u & 0x80000000u) ? ~u : (u | 0x80000000u);
}
__device__ __forceinline__ float dec_f(unsigned u) {
    unsigned v = (u & 0x80000000u) ? (u & 0x7FFFFFFFu) : ~u;
    return __uint_as_float(v);
}
__device__ __forceinline__ float lrelu(float x) { return x > 0.f ? x : NEG_SLOPE * x; }

// ---------------------------------------------------------------------------
// Kernel 1: h = x @ W via V_WMMA_F32_16X16X4_F32.
// One wave computes a 16-row slab of h (16 x 64). K=64 -> 16 wmma steps per
// 16x16 output tile; 4 tiles (one per head). A chunks held in 16 v2f regs.
// Full-tile fast path keeps the 32 result stores unconditional (no per-store
// exec save/restore; N % 16 == 0 in practice so this path always runs).
// ---------------------------------------------------------------------------
__global__ void __launch_bounds__(256) gat_gemm_h(const float* __restrict__ x,
                                                  const float* __restrict__ W,
                                                  float* __restrict__ h,
                                                  int n, int n_tiles) {
    int wave = threadIdx.x >> 5;
    int lane = threadIdx.x & 31;
    int tile = blockIdx.x * 8 + wave;
    if (tile >= n_tiles) return;          // wave-uniform: EXEC stays all-ones
    int row0 = tile << 4;
    int r  = lane & 15;                   // row within tile (A) / col within tile (B,C)
    int kh = lane >> 4;                   // K-half selector (0: K0/K1, 1: K2/K3)

    int rowA = row0 + r; if (rowA > n - 1) rowA = n - 1;   // clamp (no-op when N%16==0)
    const float* xr = x + (size_t)rowA * IN_CH + kh * 2;

    v2f A[16];
#pragma unroll
    for (int kk = 0; kk < 16; ++kk) {
        A[kk].x = xr[kk * 4 + 0];
        A[kk].y = xr[kk * 4 + 1];
    }

    const bool full = (row0 + 16 <= n);   // wave-uniform

#pragma unroll
    for (int j = 0; j < 4; ++j) {         // 4 col tiles == 4 heads
        v8f c = {};
#pragma unroll
        for (int kk = 0; kk < 16; ++kk) {
            const float* wp = W + (size_t)(kk * 4 + 2 * kh) * 64 + j * 16 + r;
            v2f b;
            b.x = wp[0];
            b.y = wp[64];
            c = __builtin_amdgcn_wmma_f32_16x16x4_f32(
                    false, A[kk], false, b, (short)0, c, false, false);
        }
        // C layout: VGPR v -> row v (lanes 0-15) / row v+8 (lanes 16-31), col = r
        float* hp = h + (size_t)row0 * 64 + j * 16 + r;
        if (full) {
#pragma unroll
            for (int v = 0; v < 8; ++v) {
                hp[(size_t)(v + 8 * kh) * 64] = c[v];
            }
        } else {
#pragma unroll
            for (int v = 0; v < 8; ++v) {
                int rr = v + 8 * kh;
                if (row0 + rr < n) hp[(size_t)rr * 64] = c[v];
            }
        }
    }
}

// ---------------------------------------------------------------------------
// Kernel 2: per-node attention coefficients + buffer init (float4 streams).
// al_src/al_dst dots, max-buffer seeded with the self-loop logit (self-loops
// are always valid, so invalid edges can be skipped in the max pass),
// denom = 0, out accumulator = 0.
// ---------------------------------------------------------------------------
__global__ void gat_node_init(const float* __restrict__ h,
                              const float* __restrict__ a_src,
                              const float* __restrict__ a_dst,
                              float* __restrict__ al_s,
                              float* __restrict__ al_d,
                              unsigned* __restrict__ maxbuf,
                              float* __restrict__ denom,
                              float* __restrict__ out_acc,
                              int n) {
    int i = blockIdx.x * blockDim.x + threadIdx.x;
    if (i >= n) return;
    const float4* hn = (const float4*)(h + (size_t)i * 64);
    const float4* as4 = (const float4*)a_src;
    const float4* ad4 = (const float4*)a_dst;
#pragma unroll
    for (int hh = 0; hh < HEADS; ++hh) {
        float s = 0.f, d = 0.f;
#pragma unroll
        for (int q = 0; q < 4; ++q) {          // 16 ch = 4 float4
            float4 v = hn[hh * 4 + q];
            float4 a = as4[hh * 4 + q];
            float4 b = ad4[hh * 4 + q];
            s += v.x * a.x + v.y * a.y + v.z * a.z + v.w * a.w;
            d += v.x * b.x + v.y * b.y + v.z * b.z + v.w * b.w;
        }
        al_s[i * 4 + hh] = s;
        al_d[i * 4 + hh] = d;
        maxbuf[i * 4 + hh] = enc_f(lrelu(s + d));   // self-loop logit
        denom[i * 4 + hh] = 0.f;
    }
    float4* oa = (float4*)(out_acc + (size_t)i * 64);
    float4 z = {0.f, 0.f, 0.f, 0.f};
#pragma unroll
    for (int q = 0; q < 16; ++q) oa[q] = z;
}

// ---------------------------------------------------------------------------
// Kernel 3: segment-max of logits over destinations (valid edges only).
// Order-preserving uint encoding -> unsigned atomicMax.
// ---------------------------------------------------------------------------
__global__ void gat_edge_max(const int* __restrict__ ei,
                             const float* __restrict__ ew,
                             const int* __restrict__ mod,
                             const float* __restrict__ al_s,
                             const float* __restrict__ al_d,
                             unsigned* __restrict__ maxbuf,
                             int ne) {
    int e = blockIdx.x * blockDim.x + threadIdx.x;
    if (e >= ne) return;
    int s = ei[e], d = ei[ne + e];
    if (mod[s] != mod[d] || ew[e] == 0.f) return;
    float4 als = *(const float4*)(al_s + (size_t)s * 4);
    float4 ald = *(const float4*)(al_d + (size_t)d * 4);
    atomicMax(&maxbuf[d * 4 + 0], enc_f(lrelu(als.x + ald.x)));
    atomicMax(&maxbuf[d * 4 + 1], enc_f(lrelu(als.y + ald.y)));
    atomicMax(&maxbuf[d * 4 + 2], enc_f(lrelu(als.z + ald.z)));
    atomicMax(&maxbuf[d * 4 + 3], enc_f(lrelu(als.w + ald.w)));
}

// ---------------------------------------------------------------------------
// Kernel 4: accumulate unnormalized softmax numerator and denominator.
//   denom[d,h]  += e
//   out[d,h,:]  += e * h[s,h,:]
// h / out / denom all L2-resident (~60 MB << 192 MB L2).
// ---------------------------------------------------------------------------
__global__ void gat_edge_acc(const int* __restrict__ ei,
                             const float* __restrict__ ew,
                             const int* __restrict__ mod,
                             const float* __restrict__ al_s,
                             const float* __restrict__ al_d,
                             const unsigned* __restrict__ maxbuf,
                             const float* __restrict__ h,
                             float* __restrict__ denom,
                             float* __restrict__ out_acc,
                             int ne) {
    int e = blockIdx.x * blockDim.x + threadIdx.x;
    if (e >= ne) return;
    int s = ei[e], d = ei[ne + e];
    if (mod[s] != mod[d] || ew[e] == 0.f) return;
    const float* hs = h + (size_t)s * 64;
    float* od = out_acc + (size_t)d * 64;
    float4 als = *(const float4*)(al_s + (size_t)s * 4);
    float4 ald = *(const float4*)(al_d + (size_t)d * 4);
    float lg[4] = { lrelu(als.x + ald.x), lrelu(als.y + ald.y),
                    lrelu(als.z + ald.z), lrelu(als.w + ald.w) };
#pragma unroll
    for (int hh = 0; hh < HEADS; ++hh) {
        float m  = dec_f(maxbuf[d * 4 + hh]);
        float ex = __expf(lg[hh] - m);
        atomicAdd(&denom[d * 4 + hh], ex);
#pragma unroll
        for (int c = 0; c < OUT_CH; c += 4) {
            float4 hv = *(const float4*)(hs + hh * 16 + c);
            atomicAdd(&od[hh * 16 + c + 0], ex * hv.x);
            atomicAdd(&od[hh * 16 + c + 1], ex * hv.y);
            atomicAdd(&od[hh * 16 + c + 2], ex * hv.z);
            atomicAdd(&od[hh * 16 + c + 3], ex * hv.w);
        }
    }
}

// ---------------------------------------------------------------------------
// Kernel 5: fold in self-loop, normalize, add bias (in place on d_out).
// ---------------------------------------------------------------------------
__global__ void gat_finalize(const float* __restrict__ h,
                             const float* __restrict__ al_s,
                             const float* __restrict__ al_d,
                             const unsigned* __restrict__ maxbuf,
                             const float* __restrict__ denom,
                             const float* __restrict__ bias,
                             float* __restrict__ out,
                             int n) {
    int i = blockIdx.x * blockDim.x + threadIdx.x;
    if (i >= n) return;
    const float4* hn = (const float4*)(h + (size_t)i * 64);
    float4* on = (float4*)(out + (size_t)i * 64);
    const float4* b4 = (const float4*)bias;
#pragma unroll
    for (int hh = 0; hh < HEADS; ++hh) {
        float m  = dec_f(maxbuf[i * 4 + hh]);
        float es = __expf(lrelu(al_s[i * 4 + hh] + al_d[i * 4 + hh]) - m);
        float dn = denom[i * 4 + hh] + es;
        float inv = 1.f / fmaxf(dn, 1e-16f);
#pragma unroll
        for (int q = 0; q < 4; ++q) {
            float4 o = on[hh * 4 + q];
            float4 hv = hn[hh * 4 + q];
            float4 bb = b4[hh * 4 + q];
            o.x = (o.x + es * hv.x) * inv + bb.x;
            o.y = (o.y + es * hv.y) * inv + bb.y;
            o.z = (o.z + es * hv.z) * inv + bb.z;
            o.w = (o.w + es * hv.w) * inv + bb.w;
            on[hh * 4 + q] = o;
        }
    }
}

extern "C" void kernel_launch(void* const* d_in, const int* in_sizes, int n_in,
                              void* d_out, int out_size, void* d_ws, size_t ws_size,
                              hipStream_t stream) {
    const float* x     = (const float*)d_in[0];
    const int*   ei    = (const int*)d_in[1];
    const float* ew    = (const float*)d_in[2];
    const int*   mod   = (const int*)d_in[3];
    const float* W     = (const float*)d_in[4];
    const float* a_src = (const float*)d_in[5];
    const float* a_dst = (const float*)d_in[6];
    const float* bias  = (const float*)d_in[7];
    float* out = (float*)d_out;

    int n  = in_sizes[0] / IN_CH;   // 100000
    int ne = in_sizes[2];           // 1600000

    // Workspace layout (floats): h[n*64] | al_s[n*4] | al_d[n*4] | max[n*4] | denom[n*4]
    float*    ws     = (float*)d_ws;
    float*    h      = ws;
    float*    al_s   = h + (size_t)n * 64;
    float*    al_d   = al_s + (size_t)n * 4;
    unsigned* maxbuf = (unsigned*)(al_d + (size_t)n * 4);
    float*    denom  = (float*)(maxbuf + (size_t)n * 4);

    int n_tiles = (n + 15) / 16;
    gat_gemm_h<<<(n_tiles + 7) / 8, 256, 0, stream>>>(x, W, h, n, n_tiles);
    gat_node_init<<<(n + 255) / 256, 256, 0, stream>>>(h, a_src, a_dst, al_s, al_d,
                                                       maxbuf, denom, out, n);
    gat_edge_max<<<(ne + 255) / 256, 256, 0, stream>>>(ei, ew, mod, al_s, al_d, maxbuf, ne);
    gat_edge_acc<<<(ne + 255) / 256, 256, 0, stream>>>(ei, ew, mod, al_s, al_d, maxbuf,
                                                       h, denom, out, ne);
    gat_finalize<<<(n + 255) / 256, 256, 0, stream>>>(h, al_s, al_d, maxbuf, denom,
                                                      bias, out, n);
}